// RocAucLoss_10874857193862
// MI455X (gfx1250) — compile-verified
//
#include <hip/hip_runtime.h>
#include <hip/hip_bf16.h>

typedef float v2f __attribute__((ext_vector_type(2)));
typedef float v8f __attribute__((ext_vector_type(8)));

#define NH      128
#define NSTEPS  1024
#define NSAMP   262144
#define NBLOCKS 256
#define CHUNK   (NSAMP / NBLOCKS)   // 1024 samples per block

// Hardware trig takes revolutions: v_sin(fract(x)) == sin(2*pi*x), exactly periodic.
__device__ __forceinline__ float sin2pi(float x) {
    return __builtin_amdgcn_sinf(__builtin_amdgcn_fractf(x));
}
__device__ __forceinline__ float cos2pi(float x) {
    return __builtin_amdgcn_cosf(__builtin_amdgcn_fractf(x));
}

__global__ void rocauc_zero_ws(float* __restrict__ ws) {
    int i = blockIdx.x * blockDim.x + threadIdx.x;
    if (i < 513) ws[i] = 0.0f;
}

// ws layout (floats):
//   [  0..127] pos sin sums   [128..255] pos cos sums
//   [256..383] neg sin sums   [384..511] neg cos sums
//   [512]      positive count
__global__ __launch_bounds__(256) void rocauc_fourier_reduce(
        const float* __restrict__ scores,
        const int*   __restrict__ targets,
        float*       __restrict__ ws) {
    const int lane   = threadIdx.x & 31;
    const int wave   = threadIdx.x >> 5;     // 0..7 -> which 16-harmonic tile
    const int mbase  = wave * 16;
    const int chunk0 = blockIdx.x * CHUNK;

    // ---- positive count: each wave counts its 1/8 slice of the chunk ----
    {
        const int c0 = chunk0 + wave * (CHUNK / 8);
        unsigned cnt = 0;
        for (int i = lane; i < CHUNK / 8; i += 32) {
            unsigned long long b = __ballot(targets[c0 + i] != 0);
            cnt += (unsigned)__popcll(b);
        }
        if (lane == 0) atomicAdd(&ws[512], (float)cnt);
    }

    // ---- WMMA masked Fourier reduction ----
    // A (16x4 f32): lane holds M = lane&15, K = (lane<16 ? {0,1} : {2,3})
    // B (4x16 f32): lane holds N = lane&15, same K rows; col0=pos mask, col1=neg mask
    const int   half = (lane < 16) ? 0 : 2;
    const int   ncol = lane & 15;
    const float hm   = (float)(mbase + (lane & 15) + 1);  // harmonic 1..128
    // b = fma(p, dsel, sneg): col0 -> p, col1 -> 1-p, others -> 0
    const float dsel = (ncol == 0) ? 1.0f : ((ncol == 1) ? -1.0f : 0.0f);
    const float sneg = (ncol == 1) ? 1.0f : 0.0f;

    v8f acc_s = {0.f,0.f,0.f,0.f,0.f,0.f,0.f,0.f};
    v8f acc_c = {0.f,0.f,0.f,0.f,0.f,0.f,0.f,0.f};

    const float2* s2 = (const float2*)scores;
    const int2*   t2 = (const int2*)targets;

    for (int g = 0; g < CHUNK; g += 4) {
        const int pair = (chunk0 + g + half) >> 1;     // 8-byte aligned pair index
        const float2 s = s2[pair];
        const int2   t = t2[pair];

        v2f a_s, a_c, b;
        a_s.x = sin2pi(hm * s.x);  a_s.y = sin2pi(hm * s.y);
        a_c.x = cos2pi(hm * s.x);  a_c.y = cos2pi(hm * s.y);

        const float p0 = (t.x != 0) ? 1.0f : 0.0f;
        const float p1 = (t.y != 0) ? 1.0f : 0.0f;
        b.x = __builtin_fmaf(p0, dsel, sneg);
        b.y = __builtin_fmaf(p1, dsel, sneg);

        acc_s = __builtin_amdgcn_wmma_f32_16x16x4_f32(
            false, a_s, false, b, (short)0, acc_s, false, false);
        acc_c = __builtin_amdgcn_wmma_f32_16x16x4_f32(
            false, a_c, false, b, (short)0, acc_c, false, false);
    }

    // D layout: VGPR j -> (M=j, N=lane) lanes 0-15; (M=j+8, N=lane-16) lanes 16-31
    if ((lane & 15) < 2) {
        const int cls  = lane & 15;               // 0 = pos, 1 = neg
        const int moff = (lane < 16) ? 0 : 8;
        float* base = ws + cls * 256;
        #pragma unroll
        for (int j = 0; j < 8; ++j) {
            const int midx = mbase + j + moff;
            atomicAdd(&base[midx],       acc_s[j]);
            atomicAdd(&base[128 + midx], acc_c[j]);
        }
    }
}

__global__ __launch_bounds__(1024) void rocauc_finalize(
        const float* __restrict__ ws, float* __restrict__ out) {
    __shared__ float isin[2][NH];
    __shared__ float icos[2][NH];
    __shared__ float pn[NSTEPS + 2];
    __shared__ float nn[NSTEPS + 2];
    __shared__ float acc[2];

    const int   tid  = threadIdx.x;
    const float eps  = 1.1920929e-07f;
    const float npos = ws[512];
    const float nneg = (float)NSAMP - npos;

    // normalized coefficients -> integrated coefficients (int_sin, int_cos)
    if (tid < 256) {
        const int   cls = tid >> 7;        // 0 pos, 1 neg
        const int   k   = tid & 127;
        const float cnt = cls ? nneg : npos;
        const float inv = (cnt < eps) ? 0.0f : (1.0f / fmaxf(cnt, eps));
        const float sc  = ws[cls * 256 + k]       * inv;   // sin coeff
        const float cc  = ws[cls * 256 + 128 + k] * inv;   // cos coeff
        const float w   = 6.28318530717958647692f * (float)(k + 1);
        isin[cls][k] =  cc / w;
        icos[cls][k] = -sc / w;
    }
    if (tid == 0) { acc[0] = 0.0f; acc[1] = 0.0f; }
    __syncthreads();

    // normalized CDFs at NSTEPS thresholds: [1, 0.75 - 0.5*cdf, 0]
    {
        const float thr = (float)tid * (1.0f / (float)(NSTEPS - 1));
        float cp = 0.0f, cn = 0.0f;
        #pragma unroll 4
        for (int k = 0; k < NH; ++k) {
            const float ang = thr * (float)(k + 1);   // revolutions
            const float s = sin2pi(ang);
            const float c = cos2pi(ang);
            cp += s * isin[0][k] + c * icos[0][k];
            cn += s * isin[1][k] + c * icos[1][k];
        }
        pn[tid + 1] = 0.75f - 0.5f * cp;
        nn[tid + 1] = 0.75f - 0.5f * cn;
        if (tid == 0) {
            pn[0] = 1.0f;          nn[0] = 1.0f;
            pn[NSTEPS + 1] = 0.0f; nn[NSTEPS + 1] = 0.0f;
        }
    }
    __syncthreads();

    // pos_auc = sum pn[i]*(nn[i]-nn[i+1]); neg_auc = 1 + sum nn[i]*(pn[i+1]-pn[i])
    float tp = 0.0f, tn = 0.0f;
    for (int i = tid; i < NSTEPS + 1; i += 1024) {
        tp += pn[i] * (nn[i] - nn[i + 1]);
        tn += nn[i] * (pn[i + 1] - pn[i]);
    }
    #pragma unroll
    for (int off = 16; off > 0; off >>= 1) {
        tp += __shfl_down(tp, off);
        tn += __shfl_down(tn, off);
    }
    if ((tid & 31) == 0) {
        atomicAdd(&acc[0], tp);
        atomicAdd(&acc[1], tn);
    }
    __syncthreads();
    if (tid == 0) out[0] = 0.5f * (acc[0] + (1.0f + acc[1]));
}

extern "C" void kernel_launch(void* const* d_in, const int* in_sizes, int n_in,
                              void* d_out, int out_size, void* d_ws, size_t ws_size,
                              hipStream_t stream) {
    const float* scores  = (const float*)d_in[0];
    const int*   targets = (const int*)d_in[1];
    float*       ws      = (float*)d_ws;
    float*       out     = (float*)d_out;

    hipLaunchKernelGGL(rocauc_zero_ws, dim3((513 + 255) / 256), dim3(256), 0, stream, ws);
    hipLaunchKernelGGL(rocauc_fourier_reduce, dim3(NBLOCKS), dim3(256), 0, stream,
                       scores, targets, ws);
    hipLaunchKernelGGL(rocauc_finalize, dim3(1), dim3(1024), 0, stream, ws, out);
}